// SharedClassName_65481071395397
// MI455X (gfx1250) — compile-verified
//
#include <hip/hip_runtime.h>

typedef __attribute__((ext_vector_type(2))) float v2f;
typedef __attribute__((ext_vector_type(8))) float v8f;

#define QS_THREADS 128   // 4 wave32s

// CNOT ladder permutation on a nibble (bit3 = group MSB):
// CNOT(q,q+1) for q=0..2 : if bit(3-q) set, flip bit(2-q)
__device__ __forceinline__ int perm4(int x) {
  if (x & 8) x ^= 4;
  if (x & 4) x ^= 2;
  if (x & 2) x ^= 1;
  return x;
}
__device__ __forceinline__ int perm4inv(int x) {
  if (x & 2) x ^= 1;
  if (x & 4) x ^= 2;
  if (x & 8) x ^= 4;
  return x;
}

__global__ __launch_bounds__(QS_THREADS) void qsim_wmma_kernel(
    const float* __restrict__ state_batch,  // (B,12)
    const float* __restrict__ params,       // (12)
    const float* __restrict__ head_w,       // (12)
    const float* __restrict__ head_b,       // (1)
    float* __restrict__ out) {
  __shared__ float psi[4096];     // one sample's statevector (16 KB)
  __shared__ float U[5][256];     // 0:B0=G0c*R0  1:R0  2:G1c  3:R1  4:M2=R2*G2c
  __shared__ float cz[24], sz[24];// cos/sin(theta/2): [0..11]=sample, [12..23]=params
  __shared__ float hws[12];
  __shared__ float red[QS_THREADS];

  const int b   = blockIdx.x;
  const int tid = threadIdx.x;

  if (tid < 12) {
    float a = 0.5f * state_batch[(size_t)b * 12 + tid];
    cz[tid] = cosf(a); sz[tid] = sinf(a);
    hws[tid] = head_w[tid];
  } else if (tid < 24) {
    float a = 0.5f * params[tid - 12];
    cz[tid] = cosf(a); sz[tid] = sinf(a);
  }
  __syncthreads();

  // Kronecker product of 4 RY(2x2) matrices; base indexes cz/sz; q=0 is nibble MSB.
  auto kron4 = [&](int base, int i, int j) -> float {
    float v = 1.0f;
#pragma unroll
    for (int q = 0; q < 4; ++q) {
      int bi = (i >> (3 - q)) & 1;
      int bj = (j >> (3 - q)) & 1;
      float c = cz[base + q], s = sz[base + q];
      v *= bi ? (bj ? c : s) : (bj ? -s : c);   // RY = [[c,-s],[s,c]]
    }
    return v;
  };

  // Build the 5 constant-per-sample 16x16 gate matrices (2 entries/thread/matrix).
  for (int e = tid; e < 5 * 256; e += QS_THREADS) {
    int mi = e >> 8, idx = e & 255, i = idx >> 4, j = idx & 15;
    float v;
    switch (mi) {
      case 0:  v = kron4(12, perm4inv(i), j); break;       // B0 = G0c * R0
      case 1:  v = kron4(12, i, j); break;                 // R0 (final layer)
      case 2:  v = (i == perm4(j)) ? 1.0f : 0.0f; break;   // G1c (permutation)
      case 3:  v = kron4(16, i, j); break;                 // R1
      default: v = kron4(20, i, perm4(j)); break;          // M2 = R2 * G2c
    }
    U[mi][idx] = v;
  }

  // psi = G0c(0) * (encoding product state).  RY|0> = [c, s]^T per wire;
  // G0c permutes the top nibble: psi[y] = prod(y with top <- perm4inv(top)).
  for (int s = tid; s < 4096; s += QS_THREADS) {
    int s2 = (perm4inv(s >> 8) << 8) | (s & 255);
    float v = 1.0f;
#pragma unroll
    for (int w = 0; w < 12; ++w)
      v *= ((s2 >> (11 - w)) & 1) ? sz[w] : cz[w];
    psi[s] = v;
  }
  __syncthreads();

  // Apply 16x16 matrix Um to the state via 16 independent tiles (4 per wave),
  // each tile = 4 chained v_wmma_f32_16x16x4_f32.  Element (k,n) of tile t
  // lives at psi[sm*k + sn*n + st*t]; in-place safe (read set == write set).
  auto apply = [&](const float* Um, int sm, int sn, int st) {
    const int lane = tid & 31, wave = tid >> 5;
    const int n = lane & 15, hi = lane >> 4;
    // A fragments: 16x4 f32 chunks of Um. Lane = row M; VGPR v holds K = 4*kc + v + 2*hi.
    v2f a[4];
#pragma unroll
    for (int kc = 0; kc < 4; ++kc) {
      int k0 = 4 * kc + 2 * hi;
      a[kc].x = Um[n * 16 + k0];
      a[kc].y = Um[n * 16 + k0 + 1];
    }
#pragma unroll
    for (int tt = 0; tt < 4; ++tt) {
      int t = wave * 4 + tt;
      int base = st * t + sn * n;
      v8f c = {};
#pragma unroll
      for (int kc = 0; kc < 4; ++kc) {
        int k0 = 4 * kc + 2 * hi;
        v2f bf;
        bf.x = psi[base + sm * k0];
        bf.y = psi[base + sm * (k0 + 1)];
        c = __builtin_amdgcn_wmma_f32_16x16x4_f32(
            /*neg_a=*/false, a[kc], /*neg_b=*/false, bf,
            /*c_mod=*/(short)0, c, /*reuse_a=*/false, /*reuse_b=*/false);
      }
#pragma unroll
      for (int v = 0; v < 8; ++v)          // D row M = v + 8*hi
        psi[base + sm * (v + 8 * hi)] = c[v];
    }
  };

  // Cross-nibble CNOT: control bit cpos, target bit cpos-1 (bit8->7, bit4->3).
  auto cswap = [&](int cpos) {
    int tpos = cpos - 1;
    for (int idx = tid; idx < 1024; idx += QS_THREADS) {
      int low = idx & ((1 << tpos) - 1);
      int high = idx >> tpos;
      int s = (high << (cpos + 1)) | (1 << cpos) | low;
      int s2 = s | (1 << tpos);
      float x = psi[s], y = psi[s2];
      psi[s] = y; psi[s2] = x;
    }
  };

  // 4 variational layers: X34; (G0c*R0 | final R0); G1c; X78; (R2*G2c); R1
  for (int l = 0; l < 4; ++l) {
    cswap(8);                       __syncthreads();
    apply(U[l < 3 ? 0 : 1], 256, 1, 16);  __syncthreads();  // top nibble
    apply(U[2], 16, 1, 256);        __syncthreads();        // middle nibble
    cswap(4);                       __syncthreads();
    apply(U[4], 1, 16, 256);        __syncthreads();        // low nibble (transposed)
    apply(U[3], 16, 1, 256);        __syncthreads();        // middle nibble
  }

  // out[b] = sum_s psi[s]^2 * sum_w hw[w]*(1 - 2*bit_w(s)) + hb
  float partial = 0.0f;
  for (int s = tid; s < 4096; s += QS_THREADS) {
    float p = psi[s]; p *= p;
    float g = 0.0f;
#pragma unroll
    for (int w = 0; w < 12; ++w)
      g += ((s >> (11 - w)) & 1) ? -hws[w] : hws[w];
    partial += p * g;
  }
  red[tid] = partial;
  __syncthreads();
#pragma unroll
  for (int off = QS_THREADS / 2; off > 0; off >>= 1) {
    if (tid < off) red[tid] += red[tid + off];
    __syncthreads();
  }
  if (tid == 0) out[b] = red[0] + head_b[0];
}

extern "C" void kernel_launch(void* const* d_in, const int* in_sizes, int n_in,
                              void* d_out, int out_size, void* d_ws, size_t ws_size,
                              hipStream_t stream) {
  const float* sb     = (const float*)d_in[0];  // state_batch (B,12)
  const float* params = (const float*)d_in[1];  // (12)
  const float* hw     = (const float*)d_in[2];  // head_w (1,12)
  const float* hb     = (const float*)d_in[3];  // head_b (1)
  float* out = (float*)d_out;
  int batch = in_sizes[0] / 12;
  qsim_wmma_kernel<<<dim3(batch), dim3(QS_THREADS), 0, stream>>>(
      sb, params, hw, hb, out);
}